// GraphConvolution_41523743818235
// MI455X (gfx1250) — compile-verified
//
#include <hip/hip_runtime.h>

#define N_NODES 50000
#define N_EDGES 800000
#define D       128   // D_IN == D_OUT

typedef __attribute__((ext_vector_type(2))) float v2f;
typedef __attribute__((ext_vector_type(4))) float v4f;
typedef __attribute__((ext_vector_type(8))) float v8f;

// ---------------- workspace layout (bytes, 64-aligned chunks) --------------
#define AGGR_BYTES   (N_NODES * D * 4)            // 25,600,000
#define CNT_OFF      (AGGR_BYTES)                 // counts : N_NODES i32
#define OFFS_OFF     (CNT_OFF + 200000)           // offsets: N_NODES+1 i32
#define CURS_OFF     (OFFS_OFF + 200064)          // cursor : N_NODES i32
#define CSR_OFF      (CURS_OFF + 200000)          // csr_src: N_EDGES i32
#define WS_NEEDED    (CSR_OFF + (size_t)N_EDGES * 4)   // ~29.4 MB

// ===========================================================================
// CSR path (no float atomics)
// ===========================================================================

// counts = 0
__global__ void __launch_bounds__(256) gc_zero_counts(int* __restrict__ counts) {
    int i = blockIdx.x * 256 + threadIdx.x;
    if (i < N_NODES) counts[i] = 0;
}

// counts[dst]++  (800K u32 atomics)
__global__ void __launch_bounds__(256) gc_count(
    const long long* __restrict__ edst, int* __restrict__ counts) {
    int e = blockIdx.x * 256 + threadIdx.x;      // grid covers exactly N_EDGES
    atomicAdd(&counts[(int)edst[e]], 1);
}

// Exclusive scan of counts -> offsets[0..N_NODES], plus cursor = offsets copy.
// Single 256-thread block, chunked Hillis-Steele scan in LDS.
__global__ void __launch_bounds__(256) gc_scan(
    const int* __restrict__ counts,
    int* __restrict__ offsets,
    int* __restrict__ cursor) {
    __shared__ int buf[256];
    __shared__ int carry;
    if (threadIdx.x == 0) carry = 0;
    __syncthreads();

    for (int base = 0; base < N_NODES; base += 256) {
        int i = base + threadIdx.x;
        int v = (i < N_NODES) ? counts[i] : 0;
        buf[threadIdx.x] = v;
        __syncthreads();
#pragma unroll
        for (int off = 1; off < 256; off <<= 1) {
            int t = (threadIdx.x >= off) ? buf[threadIdx.x - off] : 0;
            __syncthreads();
            buf[threadIdx.x] += t;
            __syncthreads();
        }
        int excl = buf[threadIdx.x] - v;         // exclusive within chunk
        if (i < N_NODES) {
            int o = carry + excl;
            offsets[i] = o;
            cursor[i]  = o;
        }
        __syncthreads();                          // all carry-reads done
        if (threadIdx.x == 255) carry += buf[255];
        __syncthreads();
    }
    if (threadIdx.x == 0) offsets[N_NODES] = carry;   // == N_EDGES
}

// csr_src[ cursor[dst]++ ] = src (cursor starts at offsets)   (800K u32 atomics)
__global__ void __launch_bounds__(256) gc_place(
    const long long* __restrict__ esrc,
    const long long* __restrict__ edst,
    int* __restrict__ cursor,
    int* __restrict__ csr_src) {
    int e   = blockIdx.x * 256 + threadIdx.x;    // grid covers exactly N_EDGES
    int d   = (int)edst[e];
    int pos = atomicAdd(&cursor[d], 1);
    csr_src[pos] = (int)esrc[e];
}

// One wave per node; lane owns 4 columns.  Pure gather + register accumulate,
// streamed store.  Zero float atomics; aggr fully overwritten (no pre-zero).
__global__ void __launch_bounds__(256) gc_gather(
    const float* __restrict__ feat,
    const int*   __restrict__ offsets,
    const int*   __restrict__ csr_src,
    float*       __restrict__ aggr) {
    int node = blockIdx.x * 8 + (threadIdx.x >> 5);   // 6250*8 == N_NODES
    int lane = threadIdx.x & 31;
    int c4   = lane * 4;

    int beg = offsets[node];
    int end = offsets[node + 1];

    v4f a0 = {0.f, 0.f, 0.f, 0.f};
    v4f a1 = {0.f, 0.f, 0.f, 0.f};
    int e = beg;
    for (; e + 2 <= end; e += 2) {                    // 2 chains hide latency
        int s0 = csr_src[e];
        int s1 = csr_src[e + 1];
        a0 += *(const v4f*)(feat + (size_t)s0 * D + c4);
        a1 += *(const v4f*)(feat + (size_t)s1 * D + c4);
    }
    if (e < end) {
        int s0 = csr_src[e];
        a0 += *(const v4f*)(feat + (size_t)s0 * D + c4);
    }
    v4f a = a0 + a1;
    *(v4f*)(aggr + (size_t)node * D + c4) = a;
}

// ===========================================================================
// Fallback path (atomic scatter) — used only if ws_size < WS_NEEDED
// ===========================================================================

__global__ void __launch_bounds__(256) gc_zero(float* __restrict__ aggr) {
    int i = blockIdx.x * 256 + threadIdx.x;
    v4f z = {0.f, 0.f, 0.f, 0.f};
    ((v4f*)aggr)[i] = z;
}

__global__ void __launch_bounds__(256) gc_scatter(
    const float*     __restrict__ feat,
    const long long* __restrict__ esrc,
    const long long* __restrict__ edst,
    float*           __restrict__ aggr) {
    int gid  = blockIdx.x * 256 + threadIdx.x;
    int edge = gid >> 5;
    int c4   = (gid & 31) * 4;

    long long s = esrc[edge];
    long long d = edst[edge];

    v4f v = *(const v4f*)(feat + s * (long long)D + c4);
    float* ap = aggr + d * (long long)D + c4;
    atomicAdd(ap + 0, v.x);
    atomicAdd(ap + 1, v.y);
    atomicAdd(ap + 2, v.z);
    atomicAdd(ap + 3, v.w);
}

// ===========================================================================
// Fused dual GEMM via V_WMMA_F32_16X16X4_F32 (unchanged from round 1 —
// assembly showed well-pipelined wmma with partial loadcnt waits).
//   out = X @ W1 + aggr @ W2 + (b1 + b2)
// One wave per 16x16 tile; 3125 blocks x 8 waves == 25000 tiles exactly,
// EXEC stays all-1s as WMMA requires.
// ===========================================================================
__global__ void __launch_bounds__(256) gc_gemm(
    const float* __restrict__ X,
    const float* __restrict__ A,
    const float* __restrict__ W1,
    const float* __restrict__ W2,
    const float* __restrict__ b1,
    const float* __restrict__ b2,
    float*       __restrict__ out) {
    const int wave  = threadIdx.x >> 5;
    const int lane  = threadIdx.x & 31;
    const int tile  = blockIdx.x * 8 + wave;
    const int tileM = tile >> 3;
    const int tileN = tile & 7;

    const int m     = lane & 15;
    const int khalf = lane >> 4;
    const int klo   = khalf * 2;
    const int col   = tileN * 16 + m;

    const float* xrow = X + (size_t)(tileM * 16 + m) * D + klo;
    const float* arow = A + (size_t)(tileM * 16 + m) * D + klo;
    const float* w1p  = W1 + (size_t)klo * D + col;
    const float* w2p  = W2 + (size_t)klo * D + col;

    v8f accX = {};
    v8f accA = {};

#pragma unroll 4
    for (int k = 0; k < D; k += 4) {
        v2f ax = *(const v2f*)(xrow + k);
        v2f aa = *(const v2f*)(arow + k);
        v2f bw1; bw1.x = w1p[(size_t)k * D]; bw1.y = w1p[(size_t)(k + 1) * D];
        v2f bw2; bw2.x = w2p[(size_t)k * D]; bw2.y = w2p[(size_t)(k + 1) * D];

        accX = __builtin_amdgcn_wmma_f32_16x16x4_f32(
            false, ax, false, bw1, (short)0, accX, false, false);
        accA = __builtin_amdgcn_wmma_f32_16x16x4_f32(
            false, aa, false, bw2, (short)0, accA, false, false);
    }

    const float bias = b1[col] + b2[col];
    float* op = out + (size_t)(tileM * 16 + khalf * 8) * D + col;
#pragma unroll
    for (int i = 0; i < 8; ++i)
        op[(size_t)i * D] = accX[i] + accA[i] + bias;
}

// ---------------------------------------------------------------------------
extern "C" void kernel_launch(void* const* d_in, const int* in_sizes, int n_in,
                              void* d_out, int out_size, void* d_ws, size_t ws_size,
                              hipStream_t stream) {
    const float*     feat = (const float*)d_in[0];
    const long long* esrc = (const long long*)d_in[1];
    const long long* edst = (const long long*)d_in[2];
    const float*     W1   = (const float*)d_in[3];
    const float*     b1   = (const float*)d_in[4];
    const float*     W2   = (const float*)d_in[5];
    const float*     b2   = (const float*)d_in[6];
    float*           out  = (float*)d_out;

    char* ws      = (char*)d_ws;
    float* aggr   = (float*)ws;
    int*   counts = (int*)(ws + CNT_OFF);
    int*   offs   = (int*)(ws + OFFS_OFF);
    int*   curs   = (int*)(ws + CURS_OFF);
    int*   csr    = (int*)(ws + CSR_OFF);

    if (ws_size >= WS_NEEDED) {
        // --- CSR path: no float atomics, pure-bandwidth aggregation ---
        gc_zero_counts<<<(N_NODES + 255) / 256, 256, 0, stream>>>(counts);
        gc_count<<<N_EDGES / 256, 256, 0, stream>>>(edst, counts);
        gc_scan<<<1, 256, 0, stream>>>(counts, offs, curs);
        gc_place<<<N_EDGES / 256, 256, 0, stream>>>(esrc, edst, curs, csr);
        gc_gather<<<N_NODES / 8, 256, 0, stream>>>(feat, offs, csr, aggr);
    } else {
        // --- fallback: float-atomic scatter (needs only aggr in ws) ---
        gc_zero<<<(N_NODES * D / 4) / 256, 256, 0, stream>>>(aggr);
        gc_scatter<<<(N_EDGES * 32) / 256, 256, 0, stream>>>(feat, esrc, edst, aggr);
    }

    // out = feat@W1 + aggr@W2 + b1 + b2   (WMMA f32 16x16x4)
    gc_gemm<<<(N_NODES / 16 * (D / 16)) / 8, 256, 0, stream>>>(
        feat, aggr, W1, W2, b1, b2, out);
}